// Decoder_8873402433674
// MI455X (gfx1250) — compile-verified
//
#include <hip/hip_runtime.h>
#include <hip/hip_bf16.h>

// ---- problem constants ----
#define BB 128
#define TT 1024
#define HH 512
#define LL 2

typedef float v2f __attribute__((ext_vector_type(2)));
typedef float v8f __attribute__((ext_vector_type(8)));

// ---- workspace layout (floats) ----
#define WS_HS      0          // 128
#define WS_ATTNAPP 128        // 128*512 = 65536
#define WS_X       65664      // 65536
#define WS_GX      131200     // 128*1536 = 196608
#define WS_GH      327808     // 196608
#define WS_PART    524416     // 128*512*8 = 524288
// total 1048704 floats (~4.2 MB)

// hs[b] = dot(hidden_attn[b,:1024], attn_w[0,:1024])
__global__ __launch_bounds__(256) void hs_kernel(const float* __restrict__ hidden,
                                                 const float* __restrict__ attn_w,
                                                 float* __restrict__ hs) {
  __shared__ float red[256];
  int b = blockIdx.x, tid = threadIdx.x;
  float acc = 0.f;
  for (int i = tid; i < LL * HH; i += 256) {
    int l = i >> 9, h = i & 511;
    acc += hidden[(size_t)l * (BB * HH) + (size_t)b * HH + h] * attn_w[i];
  }
  red[tid] = acc; __syncthreads();
  for (int s = 128; s > 0; s >>= 1) { if (tid < s) red[tid] += red[tid + s]; __syncthreads(); }
  if (tid == 0) hs[b] = red[0];
}

// score[b,t] = hs[b] + dot(enc[b,t,:], w_e) + attn_b   (1 wave per t)
__global__ __launch_bounds__(256) void score_kernel(const float* __restrict__ enc,
                                                    const float* __restrict__ attn_w,
                                                    const float* __restrict__ attn_b,
                                                    const float* __restrict__ hs,
                                                    float* __restrict__ outAttn) {
  __shared__ __align__(16) float we[HH];
  int tid = threadIdx.x;
  we[tid]       = attn_w[LL * HH + tid];
  we[tid + 256] = attn_w[LL * HH + 256 + tid];
  __syncthreads();
  int b = blockIdx.y;
  int t = blockIdx.x * 8 + (tid >> 5);
  int lane = tid & 31;
  const float4* row = (const float4*)(enc + ((size_t)b * TT + t) * HH);
  const float4* wv  = (const float4*)we;
  float acc = 0.f;
#pragma unroll
  for (int i = 0; i < 4; i++) {
    float4 e = row[i * 32 + lane];
    float4 w = wv[i * 32 + lane];
    acc += e.x * w.x + e.y * w.y + e.z * w.z + e.w * w.w;
  }
#pragma unroll
  for (int m = 16; m >= 1; m >>= 1) acc += __shfl_xor(acc, m, 32);
  if (lane == 0) outAttn[(size_t)b * TT + t] = acc + hs[b] + attn_b[0];
}

// in-place softmax over T per batch (values kept in registers between phases)
__global__ __launch_bounds__(256) void softmax_kernel(float* __restrict__ sw) {
  __shared__ float red[256];
  int b = blockIdx.x, tid = threadIdx.x;
  float* rowp = sw + (size_t)b * TT;
  float v[4];
  float mx = -INFINITY;
#pragma unroll
  for (int i = 0; i < 4; i++) { v[i] = rowp[tid + i * 256]; mx = fmaxf(mx, v[i]); }
  red[tid] = mx; __syncthreads();
  for (int s = 128; s > 0; s >>= 1) { if (tid < s) red[tid] = fmaxf(red[tid], red[tid + s]); __syncthreads(); }
  mx = red[0]; __syncthreads();
  float sum = 0.f;
#pragma unroll
  for (int i = 0; i < 4; i++) { v[i] = __expf(v[i] - mx); sum += v[i]; }
  red[tid] = sum; __syncthreads();
  for (int s = 128; s > 0; s >>= 1) { if (tid < s) red[tid] += red[tid + s]; __syncthreads(); }
  float inv = 1.f / red[0];
#pragma unroll
  for (int i = 0; i < 4; i++) rowp[tid + i * 256] = v[i] * inv;
}

// partial attn_applied over a T-chunk of 128 (grid: B x H/256 x 8)
__global__ __launch_bounds__(256) void attn_part_kernel(const float* __restrict__ enc,
                                                        const float* __restrict__ aw,
                                                        float* __restrict__ part) {
  __shared__ float wl[128];
  int b = blockIdx.x, hbase = blockIdx.y * 256, t0 = blockIdx.z * 128, tid = threadIdx.x;
  if (tid < 128) wl[tid] = aw[(size_t)b * TT + t0 + tid];
  __syncthreads();
  int h = hbase + tid;
  const float* ebase = enc + ((size_t)b * TT + t0) * HH + h;
  float acc = 0.f;
#pragma unroll 4
  for (int t = 0; t < 128; t++) acc += wl[t] * ebase[(size_t)t * HH];
  part[((size_t)b * HH + h) * 8 + blockIdx.z] = acc;
}

__global__ __launch_bounds__(256) void attn_red_kernel(const float* __restrict__ part,
                                                       float* __restrict__ app) {
  size_t idx = (size_t)blockIdx.x * 256 + threadIdx.x; // 0..65535
  float s = 0.f;
#pragma unroll
  for (int i = 0; i < 8; i++) s += part[idx * 8 + i];
  app[idx] = s;
}

// C[M x N] = A[M x K] @ W[N x (kadd+K..)]^T + bias (+ r1a[m]*r1b[n]) (+relu)
// fp32 WMMA 16x16x4, block tile 32(M) x 128(N), K-chunk 32 staged in LDS.
__global__ __launch_bounds__(256) void gemm_wmma_kernel(const float* __restrict__ A, int lda,
                                                        const float* __restrict__ W, int ldw, int kadd,
                                                        const float* __restrict__ bias,
                                                        const float* __restrict__ r1a, int r1a_stride,
                                                        const float* __restrict__ r1b, int r1b_stride,
                                                        float* __restrict__ C, int ldc,
                                                        int K, int relu) {
  __shared__ float As[32][33];
  __shared__ float Ws[128][33];
  int tid = threadIdx.x;
  int m0 = blockIdx.x * 32, n0 = blockIdx.y * 128;
  int wave = tid >> 5, lane = tid & 31;
  int wm = wave >> 2, wn = wave & 3;          // 2x4 wave grid -> 16x32 per wave
  int row = lane & 15, hi = lane >> 4, koff = hi * 2;

  v8f acc0 = {}; v8f acc1 = {};
  for (int kc = 0; kc < K; kc += 32) {
#pragma unroll
    for (int i = 0; i < 4; i++) {
      int idx = tid + i * 256;
      As[idx >> 5][idx & 31] = A[(size_t)(m0 + (idx >> 5)) * lda + kc + (idx & 31)];
    }
#pragma unroll
    for (int i = 0; i < 16; i++) {
      int idx = tid + i * 256;
      Ws[idx >> 5][idx & 31] = W[(size_t)(n0 + (idx >> 5)) * ldw + kadd + kc + (idx & 31)];
    }
    __syncthreads();
#pragma unroll
    for (int kk = 0; kk < 32; kk += 4) {
      // A frag: lanes 0-15 hold A[row][k],A[row][k+1]; lanes 16-31 hold A[row][k+2],A[row][k+3]
      v2f a;  a.x  = As[wm * 16 + row][kk + koff];      a.y  = As[wm * 16 + row][kk + koff + 1];
      // B frag mirrors A: B[k][n] = W[n][k]
      v2f b0; b0.x = Ws[wn * 32 + row][kk + koff];      b0.y = Ws[wn * 32 + row][kk + koff + 1];
      v2f b1; b1.x = Ws[wn * 32 + 16 + row][kk + koff]; b1.y = Ws[wn * 32 + 16 + row][kk + koff + 1];
      acc0 = __builtin_amdgcn_wmma_f32_16x16x4_f32(false, a, false, b0, (short)0, acc0, false, false);
      acc1 = __builtin_amdgcn_wmma_f32_16x16x4_f32(false, a, false, b1, (short)0, acc1, false, false);
    }
    __syncthreads();
  }
  // epilogue: C/D layout -> VGPR i: lanes0-15 M=i, lanes16-31 M=i+8; N=lane&15
#pragma unroll
  for (int i = 0; i < 8; i++) {
    int m  = m0 + wm * 16 + i + hi * 8;
    int na = n0 + wn * 32 + row;
    int nb = na + 16;
    float r1m = r1b ? r1a[(size_t)m * r1a_stride] : 0.f;
    float v0 = acc0[i] + bias[na] + (r1b ? r1m * r1b[(size_t)na * r1b_stride] : 0.f);
    float v1 = acc1[i] + bias[nb] + (r1b ? r1m * r1b[(size_t)nb * r1b_stride] : 0.f);
    if (relu) { v0 = fmaxf(v0, 0.f); v1 = fmaxf(v1, 0.f); }
    C[(size_t)m * ldc + na] = v0;
    C[(size_t)m * ldc + nb] = v1;
  }
}

// GRU gate fusion: r,z,n -> h_new; writes hidden_out slice and next-layer x
__global__ __launch_bounds__(256) void gates_kernel(const float* __restrict__ gx,
                                                    const float* __restrict__ gh,
                                                    const float* __restrict__ hold,
                                                    float* __restrict__ hout,
                                                    float* __restrict__ xout) {
  size_t idx = (size_t)blockIdx.x * 256 + threadIdx.x; // 0..65535
  size_t b = idx >> 9, j = idx & 511;
  size_t base = b * (3 * HH) + j;
  float xr = gx[base], xz = gx[base + HH], xn = gx[base + 2 * HH];
  float hr = gh[base], hz = gh[base + HH], hn = gh[base + 2 * HH];
  float r = 1.f / (1.f + __expf(-(xr + hr)));
  float z = 1.f / (1.f + __expf(-(xz + hz)));
  float n = tanhf(xn + r * hn);
  float ho = hold[idx];
  float hnew = (1.f - z) * n + z * ho;
  hout[idx] = hnew;
  xout[idx] = hnew;
}

// output[b] = dot(x[b,:], out_w[0,:]) + out_b[0]
__global__ __launch_bounds__(256) void out_kernel(const float* __restrict__ x,
                                                  const float* __restrict__ ow,
                                                  const float* __restrict__ ob,
                                                  float* __restrict__ out) {
  __shared__ float red[256];
  int b = blockIdx.x, tid = threadIdx.x;
  float acc = x[(size_t)b * HH + tid] * ow[tid] + x[(size_t)b * HH + 256 + tid] * ow[256 + tid];
  red[tid] = acc; __syncthreads();
  for (int s = 128; s > 0; s >>= 1) { if (tid < s) red[tid] += red[tid + s]; __syncthreads(); }
  if (tid == 0) out[b] = red[0] + ob[0];
}

extern "C" void kernel_launch(void* const* d_in, const int* in_sizes, int n_in,
                              void* d_out, int out_size, void* d_ws, size_t ws_size,
                              hipStream_t stream) {
  const float* input  = (const float*)d_in[0];   // (B,8)
  const float* hidden = (const float*)d_in[1];   // (L,B,H)
  const float* enc    = (const float*)d_in[2];   // (B,T,H)
  const float* attn_w = (const float*)d_in[3];   // (1,(L+1)*H)
  const float* attn_b = (const float*)d_in[4];   // (1,)
  const float* comb_w = (const float*)d_in[5];   // (H,H+1)
  const float* comb_b = (const float*)d_in[6];   // (H,)
  const float* w_ih   = (const float*)d_in[7];   // (L,3H,H)
  const float* w_hh   = (const float*)d_in[8];   // (L,3H,H)
  const float* b_ih   = (const float*)d_in[9];   // (L,3H)
  const float* b_hh   = (const float*)d_in[10];  // (L,3H)
  const float* out_w  = (const float*)d_in[11];  // (1,H)
  const float* out_b  = (const float*)d_in[12];  // (1,)

  float* out       = (float*)d_out;
  float* outHidden = out + BB;                       // (L,B,H)
  float* outAttn   = out + BB + LL * BB * HH;        // (B,1,T)
  float* ws        = (float*)d_ws;

  // 1) hidden score
  hs_kernel<<<BB, 256, 0, stream>>>(hidden, attn_w, ws + WS_HS);
  // 2) attention scores (enc pass 1) -> stored in d_out attn region
  score_kernel<<<dim3(TT / 8, BB), 256, 0, stream>>>(enc, attn_w, attn_b, ws + WS_HS, outAttn);
  // 3) softmax in place -> attn_weights
  softmax_kernel<<<BB, 256, 0, stream>>>(outAttn);
  // 4) attn_applied (enc pass 2): split-T partials + deterministic reduce
  attn_part_kernel<<<dim3(BB, HH / 256, 8), 256, 0, stream>>>(enc, outAttn, ws + WS_PART);
  attn_red_kernel<<<(BB * HH) / 256, 256, 0, stream>>>(ws + WS_PART, ws + WS_ATTNAPP);
  // 5) combine: x = relu([input0, attn_applied] @ comb_w.T + comb_b)
  //    K=512 over attn_applied with comb_w[:,1:]; input0 column folded in as rank-1 term.
  gemm_wmma_kernel<<<dim3(BB / 32, HH / 128), 256, 0, stream>>>(
      ws + WS_ATTNAPP, HH, comb_w, HH + 1, 1, comb_b,
      input + 7, 8, comb_w, HH + 1, ws + WS_X, HH, HH, 1);
  // 6) GRU layers
  for (int l = 0; l < LL; l++) {
    const float* wl_ih = w_ih + (size_t)l * 3 * HH * HH;
    const float* wl_hh = w_hh + (size_t)l * 3 * HH * HH;
    const float* bl_ih = b_ih + (size_t)l * 3 * HH;
    const float* bl_hh = b_hh + (size_t)l * 3 * HH;
    const float* hl    = hidden + (size_t)l * BB * HH;
    gemm_wmma_kernel<<<dim3(BB / 32, (3 * HH) / 128), 256, 0, stream>>>(
        ws + WS_X, HH, wl_ih, HH, 0, bl_ih, nullptr, 0, nullptr, 0,
        ws + WS_GX, 3 * HH, HH, 0);
    gemm_wmma_kernel<<<dim3(BB / 32, (3 * HH) / 128), 256, 0, stream>>>(
        hl, HH, wl_hh, HH, 0, bl_hh, nullptr, 0, nullptr, 0,
        ws + WS_GH, 3 * HH, HH, 0);
    gates_kernel<<<(BB * HH) / 256, 256, 0, stream>>>(
        ws + WS_GX, ws + WS_GH, hl, outHidden + (size_t)l * BB * HH, ws + WS_X);
  }
  // 7) final projection
  out_kernel<<<BB, 256, 0, stream>>>(ws + WS_X, out_w, out_b, out);
}